// Bert_Propagation_55731495632984
// MI455X (gfx1250) — compile-verified
//
#include <hip/hip_runtime.h>

#define D      128      // feature dim (fixed by reference)
#define EPB    128      // edges staged per block
#define BLOCK  256      // 8 waves (wave32)

typedef __attribute__((ext_vector_type(2))) float v2f;
typedef __attribute__((ext_vector_type(8))) float v8f;

// ---- CDNA5 async global->LDS staging (inline asm; portable across toolchains) ----
__device__ __forceinline__ unsigned lds_off_u32(void* p) {
    // generic -> LDS(3) addrspace cast, then take the 32-bit LDS byte offset
    return (unsigned)(unsigned long long)(__attribute__((address_space(3))) void*)p;
}
__device__ __forceinline__ void async_g2l_b64(unsigned lds_off, const void* g) {
    asm volatile("global_load_async_to_lds_b64 %0, %1, off"
                 :: "v"(lds_off), "v"((unsigned long long)g) : "memory");
}
__device__ __forceinline__ void async_g2l_b32(unsigned lds_off, const void* g) {
    asm volatile("global_load_async_to_lds_b32 %0, %1, off"
                 :: "v"(lds_off), "v"((unsigned long long)g) : "memory");
}
__device__ __forceinline__ void wait_async0() {
    asm volatile("s_wait_asynccnt 0" ::: "memory");
}

// ---------------- COO SpMM scatter: out[r] += v * x[c]  (row = D floats) ----------------
__global__ __launch_bounds__(BLOCK) void spmm_scatter(
    const long long* __restrict__ rows,
    const long long* __restrict__ cols,
    const float*     __restrict__ vals,
    const float*     __restrict__ x,
    float*           __restrict__ out,
    int E)
{
    __shared__ long long s_row[EPB];
    __shared__ long long s_col[EPB];
    __shared__ float     s_val[EPB];

    const int e0 = blockIdx.x * EPB;
    const int t  = threadIdx.x;

    // Stage this block's edge triples into LDS with the async DMA path.
    if (t < EPB) {
        const int e = e0 + t;
        if (e < E) {
            async_g2l_b64(lds_off_u32(&s_row[t]), &rows[e]);
            async_g2l_b64(lds_off_u32(&s_col[t]), &cols[e]);
            async_g2l_b32(lds_off_u32(&s_val[t]), &vals[e]);
        }
    }
    wait_async0();
    __syncthreads();

    const int wave = t >> 5;   // 0..7
    const int lane = t & 31;   // float4 chunk: 32 lanes * 4 floats = 128 = D

    #pragma unroll 4
    for (int j = 0; j < EPB / 8; ++j) {
        const int el = wave + (j << 3);
        const int e  = e0 + el;
        if (e >= E) continue;                    // uniform per wave
        const long long c = s_col[el];
        const long long r = s_row[el];
        const float     v = s_val[el];

        const float4 xv = *((const float4*)(x + (size_t)c * D) + lane);

        // prefetch next edge's gather row (global_prefetch_b8)
        if (j + 1 < EPB / 8) {
            const int el2 = el + 8;
            if (e0 + el2 < E) {
                const long long c2 = s_col[el2];
                __builtin_prefetch((const void*)((const float4*)(x + (size_t)c2 * D) + lane), 0, 0);
            }
        }

        float* o = out + (size_t)r * D + lane * 4;
        atomicAdd(o + 0, v * xv.x);              // global_atomic_add_f32 (no-rtn)
        atomicAdd(o + 1, v * xv.y);
        atomicAdd(o + 2, v * xv.z);
        atomicAdd(o + 3, v * xv.w);
    }
}

// ---------------- Epilogue: out = c0*X + ck*Y via chained V_WMMA_F32_16X16X4_F32 ----------------
// For each 16x16 tile: D = sum_{j=0..3} A_j * B_j + C, with A_j = X[:, tc+4j..tc+4j+3],
// B_j[k,n] = c0 * delta(n == 4j+k), C = ck * Y_tile.  Exact in fp32.
__global__ __launch_bounds__(256) void combine_wmma(
    const float* __restrict__ X,
    const float* Y,                // may alias out (each element touched by one lane only)
    const float* __restrict__ temp,
    int k, int Nrows,
    float* out)
{
    const float t0 = temp[0] > 0.f ? temp[0] : 0.f;
    const float tk = temp[k] > 0.f ? temp[k] : 0.f;
    const float scale = 1.0f / (float)(1 << k);
    const float c0 = scale * t0;
    const float ck = scale * tk;

    const int wave = blockIdx.x * (blockDim.x >> 5) + (threadIdx.x >> 5);
    const int lane = threadIdx.x & 31;
    const int tilesPerBand = D / 16;                 // 8
    const int nTiles = (Nrows / 16) * tilesPerBand;  // 50000/16 * 8 = 25000
    if (wave >= nTiles) return;                      // wave-uniform; EXEC stays all-1s

    const int tr = (wave / tilesPerBand) * 16;
    const int tc = (wave % tilesPerBand) * 16;

    const int nloc = lane & 15;       // C/D: column within tile
    const int half = lane >> 4;       // 0: rows 0..7 / K 0,1 ; 1: rows 8..15 / K 2,3

    // C = ck * Y tile (16x16 f32 layout: VGPR v -> row v + 8*half, col nloc)
    v8f c;
    #pragma unroll
    for (int v = 0; v < 8; ++v)
        c[v] = ck * Y[(size_t)(tr + v + 8 * half) * D + tc + nloc];

    #pragma unroll
    for (int j = 0; j < 4; ++j) {
        const int k0 = half * 2;
        // A (16x4 f32): lanes 0-15 row=lane K={0,1}; lanes 16-31 row=lane-16 K={2,3}
        v2f a;
        a.x = X[(size_t)(tr + nloc) * D + tc + j * 4 + k0];
        a.y = X[(size_t)(tr + nloc) * D + tc + j * 4 + k0 + 1];
        // B (4x16): mirrored layout; B[k,n] = c0 * (n == 4j+k)
        v2f b;
        b.x = (nloc == j * 4 + k0    ) ? c0 : 0.f;
        b.y = (nloc == j * 4 + k0 + 1) ? c0 : 0.f;
        c = __builtin_amdgcn_wmma_f32_16x16x4_f32(false, a, false, b,
                                                  (short)0, c, false, false);
    }

    #pragma unroll
    for (int v = 0; v < 8; ++v)
        out[(size_t)(tr + v + 8 * half) * D + tc + nloc] = c[v];
}

extern "C" void kernel_launch(void* const* d_in, const int* in_sizes, int n_in,
                              void* d_out, int out_size, void* d_ws, size_t ws_size,
                              hipStream_t stream)
{
    const float*     input     = (const float*)d_in[0];
    const long long* adj_idx   = (const long long*)d_in[1];  // [2,E] int64
    const float*     adj_vals  = (const float*)d_in[2];
    const long long* adj2_idx  = (const long long*)d_in[3];
    const float*     adj2_vals = (const float*)d_in[4];
    const float*     temp      = (const float*)d_in[5];

    const int E     = in_sizes[2];
    const int ND    = in_sizes[0];
    const int Nrows = ND / D;
    const int k     = in_sizes[5] - 1;

    float* A = (float*)d_ws;
    float* B = A + (size_t)ND;
    float* O = (float*)d_out;
    const size_t bytes = (size_t)ND * sizeof(float);

    dim3 sgrid((E + EPB - 1) / EPB), sblk(BLOCK);

    // X chain: (adj2)^k @ input, ping-pong A/B
    const float* src = input;
    for (int i = 0; i < k; ++i) {
        float* dst = (i & 1) ? B : A;
        hipMemsetAsync(dst, 0, bytes, stream);
        spmm_scatter<<<sgrid, sblk, 0, stream>>>(adj2_idx, adj2_idx + E, adj2_vals, src, dst, E);
        src = dst;
    }
    const float* xfin = src;

    // Y chain: (adj)^k @ input, ping-pong (free ws buffer) / d_out
    float* other = (xfin == A) ? B : A;
    src = input;
    for (int i = 0; i < k; ++i) {
        float* dst = (i & 1) ? O : other;
        hipMemsetAsync(dst, 0, bytes, stream);
        spmm_scatter<<<sgrid, sblk, 0, stream>>>(adj_idx, adj_idx + E, adj_vals, src, dst, E);
        src = dst;
    }
    const float* yfin = src;

    // Epilogue: out = c0*xfin + ck*yfin  (WMMA, one wave per 16x16 tile)
    const int nTiles = (Nrows / 16) * (D / 16);
    const int wpb = 256 / 32;
    dim3 cgrid((nTiles + wpb - 1) / wpb), cblk(256);
    combine_wmma<<<cgrid, cblk, 0, stream>>>(xfin, yfin, temp, k, Nrows, O);
}